// GaugeField_67405216744023
// MI455X (gfx1250) — compile-verified
//
#include <hip/hip_runtime.h>

// ---------------------------------------------------------------------------
// GaugeField transport on MI455X (gfx1250, wave32)
//
// Phase 1 (expm_kernel): U[e] = expm(0.5*(omega[e]-omega[e]^T)), K=8.
//   Two 8x8 matrices packed block-diagonally into a 16x16 per wave; each
//   16x16x16 matmul is 4 chained V_WMMA_F32_16X16X4_F32. Taylor to order 8
//   (||W|| ~ 0.05 => remainder ~5e-18, below f32 eps; no squaring needed).
//   Skew-symmetry gives the B-layout operand as just -A-layout (same lane).
// Phase 2 (apply_kernel): out[b] = U[idx[b]] @ J[b]; U (128MB) is L2-resident
//   on MI455X (192MB L2), so the random 256B/event gather is cheap.
// ---------------------------------------------------------------------------

typedef float v2f __attribute__((ext_vector_type(2)));
typedef float v4f __attribute__((ext_vector_type(4)));
typedef float v8f __attribute__((ext_vector_type(8)));

// w: 128 floats = two 8x8 omegas (edge0 = rows 0..7, edge1 = rows 8..15 of
// the packed 16x16). Branchless element (m,k) of the packed skew matrix:
// loads are always in-bounds; off-block elements select 0 via cndmask.
__device__ __forceinline__ float skew_at(const float* w, int m, int k) {
    const int base = (m & 8) ? 64 : 0;
    const int i = m & 7, j = k & 7;
    const float v = 0.5f * (w[base + i * 8 + j] - w[base + j * 8 + i]);
    return ((m >> 3) == (k >> 3)) ? v : 0.0f;
}

// D(16x16) = A(16x16) * B(16x16) as 4 chained 16x16x4 f32 WMMAs.
// a[2c+v] = A-layout chunk c (K=4c..4c+3), b[2c+v] = B-layout chunk c.
__device__ __forceinline__ v8f mm16(const float* a, const float* b) {
    v8f c = {0.f, 0.f, 0.f, 0.f, 0.f, 0.f, 0.f, 0.f};
#pragma unroll
    for (int ch = 0; ch < 4; ++ch) {
        v2f av; av.x = a[2 * ch]; av.y = a[2 * ch + 1];
        v2f bv; bv.x = b[2 * ch]; bv.y = b[2 * ch + 1];
        // 8 args: (neg_a, A, neg_b, B, c_mod, C, reuse_a, reuse_b)
        c = __builtin_amdgcn_wmma_f32_16x16x4_f32(
                false, av, false, bv, (short)0, c, false, false);
    }
    return c;
}

__global__ void __launch_bounds__(256)
expm_kernel(const float* __restrict__ omega, float* __restrict__ U, int E) {
    __shared__ float lds[8][128];           // 8 waves/block, 2 edges/wave
    const int  lane = threadIdx.x & 31;
    const int  wis  = threadIdx.x >> 5;
    const long wid  = (long)blockIdx.x * 8 + wis;

    const long emax = (long)E - 1;
    long e0 = 2 * wid;     if (e0 > emax) e0 = emax;   // clamped waves compute
    long e1 = e0 + 1;      if (e1 > emax) e1 = emax;   // identical duplicates

    // --- stage omega for both edges into LDS (coalesced float2 loads) ---
    {
        const float2* s0 = (const float2*)(omega + e0 * 64);
        const float2* s1 = (const float2*)(omega + e1 * 64);
        float2* d = (float2*)&lds[wis][0];
        d[lane]      = s0[lane];
        d[lane + 32] = s1[lane];
    }
    __syncthreads();

    const int h  = lane >> 4;     // lane half
    const int nl = lane & 15;     // M (for A-layout) / N (for B,C,D layouts)
    const float* W = &lds[wis][0];

    // --- build operands ---
    // A 16x4 chunk c: lanes 0-15 hold {K=4c,4c+1}, lanes 16-31 {K=4c+2,4c+3}
    //   wA[2c+v] (this lane) = W[nl][4c+2h+v]
    // B 4x16 chunk c: VGPR v holds K-row 4c+v (lanes 0-15) / 4c+v+2 (16-31)
    //   wB[2c+v] (this lane) = W[4c+v+2h][nl] = -W[nl][4c+2h+v] = -wA[2c+v]
    //   (identical index expression; skew-symmetry supplies the sign)
    // C/D: VGPR r holds row r (lanes 0-15) / r+8 (lanes 16-31)
    float wA[8], P[8], uCD[8];
#pragma unroll
    for (int c = 0; c < 4; ++c)
#pragma unroll
        for (int v = 0; v < 2; ++v)
            wA[2 * c + v] = skew_at(W, nl, 4 * c + 2 * h + v);
#pragma unroll
    for (int i = 0; i < 8; ++i) P[i] = -wA[i];   // P = W^1 in B-layout
#pragma unroll
    for (int r = 0; r < 8; ++r) {
        const int m = r + 8 * h;
        uCD[r] = ((m == nl) ? 1.0f : 0.0f) + skew_at(W, m, nl);  // I + W
    }

    // --- Taylor: U = I + W + W^2/2! + ... + W^8/8! ---
    const float coef[9] = {0.f, 0.f, 0.5f, 1.f / 6.f, 1.f / 24.f, 1.f / 120.f,
                           1.f / 720.f, 1.f / 5040.f, 1.f / 40320.f};
#pragma unroll
    for (int j = 2; j <= 8; ++j) {
        v8f d = mm16(wA, P);                     // d = W^j (C/D layout)
        float dv[8];
#pragma unroll
        for (int r = 0; r < 8; ++r) {
            dv[r] = d[r];
            uCD[r] += dv[r] * coef[j];
        }
        if (j < 8) {
            // C/D -> B layout: element (k = 4c+v+2h, n) sits in CD[k&7]
            // at source lane n + 16*(k>>3); k&7 is compile-time per half.
#pragma unroll
            for (int c = 0; c < 4; ++c)
#pragma unroll
                for (int v = 0; v < 2; ++v) {
                    const int klo = 4 * c + v;       // half 0
                    const int khi = 4 * c + v + 2;   // half 1
                    float lo = __shfl(dv[klo & 7], nl + 16 * (klo >> 3), 32);
                    float hi = __shfl(dv[khi & 7], nl + 16 * (khi >> 3), 32);
                    P[2 * c + v] = h ? hi : lo;
                }
        }
    }

    // --- store in-block elements of the packed result ---
    // half 0 owns edge e0 (its columns are n<8), half 1 owns edge e1 (n>=8)
    const bool inblk = (h == (nl >> 3));
    const long e     = h ? e1 : e0;
    if (inblk) {
        float* Ue = U + e * 64;
#pragma unroll
        for (int r = 0; r < 8; ++r)
            Ue[r * 8 + (nl & 7)] = uCD[r];
    }
}

__global__ void __launch_bounds__(256)
apply_kernel(const float* __restrict__ U, const float* __restrict__ J,
             const int* __restrict__ idx, float* __restrict__ out, int B) {
    const long b = (long)blockIdx.x * blockDim.x + threadIdx.x;
    if (b >= B) return;
    const long e = (long)idx[b];

    const float4* Jr = (const float4*)(J + b * 8);
    const float4 j0 = Jr[0], j1 = Jr[1];
    const float jv[8] = {j0.x, j0.y, j0.z, j0.w, j1.x, j1.y, j1.z, j1.w};

    const float* Ue = U + e * 64;      // 256B per event, L2-resident
    float o[8];
#pragma unroll
    for (int i = 0; i < 8; ++i) {
        const float4* Ur = (const float4*)(Ue + i * 8);
        const float4 u0 = Ur[0], u1 = Ur[1];
        o[i] = u0.x * jv[0] + u0.y * jv[1] + u0.z * jv[2] + u0.w * jv[3] +
               u1.x * jv[4] + u1.y * jv[5] + u1.z * jv[6] + u1.w * jv[7];
    }
    // write-once 64MB output: non-temporal b128 stores, keep L2 for U
    v4f* O = (v4f*)(out + b * 8);
    v4f o0 = {o[0], o[1], o[2], o[3]};
    v4f o1 = {o[4], o[5], o[6], o[7]};
    __builtin_nontemporal_store(o0, &O[0]);
    __builtin_nontemporal_store(o1, &O[1]);
}

extern "C" void kernel_launch(void* const* d_in, const int* in_sizes, int n_in,
                              void* d_out, int out_size, void* d_ws, size_t ws_size,
                              hipStream_t stream) {
    const float* omega = (const float*)d_in[0];   // (E,8,8) f32
    const float* J     = (const float*)d_in[1];   // (B,8)   f32
    const int*   idx   = (const int*)d_in[2];     // (B,)    int32 (per harness)
    // d_in[3] (edges) is unused by the reference output.

    const int E = in_sizes[0] / 64;
    const int B = in_sizes[1] / 8;

    float* U = (float*)d_ws;                      // E*64*4 bytes of scratch

    const int numWaves = (E + 1) / 2;             // 2 edges per wave
    const int blocks1  = (numWaves + 7) / 8;      // 8 waves per block
    expm_kernel<<<blocks1, 256, 0, stream>>>(omega, U, E);

    const int blocks2 = (B + 255) / 256;
    apply_kernel<<<blocks2, 256, 0, stream>>>(U, J, idx, (float*)d_out, B);
}